// GIN_61804579389456
// MI455X (gfx1250) — compile-verified
//
#include <hip/hip_runtime.h>
#include <hip/hip_bf16.h>

// GIN 3-layer forward for MI455X (gfx1250, wave32).
//   agg[i] = sum_{e: dst[e]==i} h[src[e]]       (scatter-add, global f32 atomics)
//   h'     = (h + agg) @ W + b                   (WMMA f32 16x16x4, W staged in LDS)
//
// N_NODES = 100000 (= 6250 * 16, no M padding needed)
// N_EDGES = 1000000, dims 128 -> 64 -> 64 -> 16
// All index arithmetic fits in 32 bits (max 100000*128 = 12.8M elements).

#define GIN_N_NODES 100000
#define GIN_N_EDGES 1000000

typedef __attribute__((ext_vector_type(2))) float v2f;
typedef __attribute__((ext_vector_type(8))) float v8f;

// ---------------------------------------------------------------------------
// Zero a float buffer (float4 stores), n4 = number of float4 elements.
// ---------------------------------------------------------------------------
__global__ void gin_zero_kernel(float4* __restrict__ p, int n4) {
  int i = blockIdx.x * blockDim.x + threadIdx.x;
  if (i < n4) {
    p[i] = make_float4(0.f, 0.f, 0.f, 0.f);
  }
}

// ---------------------------------------------------------------------------
// Scatter-add: agg[dst[e]][*] += h[src[e]][*],  D = 4 << LOG2C floats per row.
// One thread handles one float4 chunk of one edge row; shifts/masks instead of
// runtime division. For D=128 a wave32 covers exactly one 512B row (coalesced
// gather); the adds lower to global_atomic_add_f32.
// ---------------------------------------------------------------------------
template <int LOG2C>
__global__ void gin_scatter_kernel(const float* __restrict__ h,
                                   const int* __restrict__ src,
                                   const int* __restrict__ dst,
                                   float* __restrict__ agg,
                                   int nEdges) {
  const unsigned t = blockIdx.x * blockDim.x + threadIdx.x;
  if (t >= ((unsigned)nEdges << LOG2C)) return;
  const int e = (int)(t >> LOG2C);
  const int c = (int)(t & ((1u << LOG2C) - 1u));
  const int D = 4 << LOG2C;
  const int s = src[e];
  const int d = dst[e];
  const float4 v = *(const float4*)(h + s * D + (c << 2));
  float* o = agg + d * D + (c << 2);
  __hip_atomic_fetch_add(o + 0, v.x, __ATOMIC_RELAXED, __HIP_MEMORY_SCOPE_AGENT);
  __hip_atomic_fetch_add(o + 1, v.y, __ATOMIC_RELAXED, __HIP_MEMORY_SCOPE_AGENT);
  __hip_atomic_fetch_add(o + 2, v.z, __ATOMIC_RELAXED, __HIP_MEMORY_SCOPE_AGENT);
  __hip_atomic_fetch_add(o + 3, v.w, __ATOMIC_RELAXED, __HIP_MEMORY_SCOPE_AGENT);
}

// ---------------------------------------------------------------------------
// Fused GIN linear layer: Out[M,N] = (H[M,K] + Agg[M,K]) @ W[K,N] + bias[N]
//
// Grid: x = group of 8 M-tiles (8 waves / 256 threads per block)
//       y = N-tile (16 output columns)
// The block stages its W[:,n0:n0+16] slice TRANSPOSED into LDS (wTile[n*K+k])
// so the per-lane B operand pair (k, k+1) is one contiguous ds_load_b64.
//
// ISA VGPR layouts (CDNA5 ISA, V_WMMA_F32_16X16X4_F32):
//   A 16x4 f32 : lanes 0-15 -> M=lane, VGPR{0,1}=K{0,1};  lanes 16-31 K{2,3}
//   B 4x16 f32 : lanes 0-15 -> N=lane, VGPR{0,1}=K{0,1};  lanes 16-31 K{2,3}
//   C/D 16x16  : row = 8*(lane>=16) + vgpr, col = lane & 15
// __syncthreads() precedes the (wave-uniform) bounds return, so the barrier is
// never divergent and EXEC is all-1s at every WMMA.
// ---------------------------------------------------------------------------
template <int K>
__global__ void gin_gemm_kernel(const float* __restrict__ H,
                                const float* __restrict__ Agg,
                                const float* __restrict__ W,
                                const float* __restrict__ bias,
                                float* __restrict__ Out,
                                int M, int N) {
  __shared__ float wTile[16 * K];                // transposed B slice [n][k]

  const int tid   = threadIdx.x;
  const int lane  = tid & 31;
  const int n0    = blockIdx.y << 4;             // first output column

  // Cooperative staging: wTile[n*K + k] = W[k*N + n0 + n]; K is a power of 2
  // so /K and %K are shifts/masks at compile time.
#pragma unroll
  for (int idx = tid; idx < 16 * K; idx += 256) {
    const int n = idx / K;
    const int k = idx % K;
    wTile[idx] = W[k * N + n0 + n];
  }
  __syncthreads();

  const int tileM = (blockIdx.x << 3) + (tid >> 5);
  if ((tileM << 4) >= M) return;                 // wave-uniform guard

  const int l15 = lane & 15;
  const int kg  = (lane >> 4) << 1;              // K sub-offset: 0 or 2
  const int m   = (tileM << 4) + l15;            // A row for this lane

  const float* __restrict__ hRow = H   + m * K;  // 32-bit index math
  const float* __restrict__ aRow = Agg + m * K;
  const float* __restrict__ bRow = wTile + l15 * K;

  v8f acc = {};
#pragma unroll
  for (int k0 = 0; k0 < K; k0 += 4) {
    v2f a, b;
    a.x = hRow[k0 + kg + 0] + aRow[k0 + kg + 0]; // global_load_b64 x2 + pk_add
    a.y = hRow[k0 + kg + 1] + aRow[k0 + kg + 1];
    b.x = bRow[k0 + kg + 0];                     // one ds_load_b64
    b.y = bRow[k0 + kg + 1];
    // (neg_a, A, neg_b, B, c_mod, C, reuse_a, reuse_b)
    acc = __builtin_amdgcn_wmma_f32_16x16x4_f32(false, a, false, b,
                                                (short)0, acc, false, false);
  }

  const float bv = bias[n0 + l15];
  const int rowBase = (tileM << 4) + ((lane >> 4) << 3);
#pragma unroll
  for (int r = 0; r < 8; ++r) {
    Out[(rowBase + r) * N + n0 + l15] = acc[r] + bv;
  }
}

// ---------------------------------------------------------------------------
// Host-side orchestration (graph-capture safe: kernels on `stream` only).
// ---------------------------------------------------------------------------
static inline int cdiv(long long a, int b) { return (int)((a + b - 1) / b); }

static void run_zero(float* p, long long nFloats, hipStream_t stream) {
  int n4 = (int)(nFloats >> 2);
  gin_zero_kernel<<<cdiv(n4, 256), 256, 0, stream>>>((float4*)p, n4);
}

extern "C" void kernel_launch(void* const* d_in, const int* in_sizes, int n_in,
                              void* d_out, int out_size, void* d_ws, size_t ws_size,
                              hipStream_t stream) {
  const float* x   = (const float*)d_in[0];   // [100000,128]
  const int*   src = (const int*)d_in[1];     // [1000000]
  const int*   dst = (const int*)d_in[2];     // [1000000]
  const float* w1  = (const float*)d_in[3];   // [128,64]
  const float* b1  = (const float*)d_in[4];   // [64]
  const float* w2  = (const float*)d_in[5];   // [64,64]
  const float* b2  = (const float*)d_in[6];   // [64]
  const float* w3  = (const float*)d_in[7];   // [64,16]
  const float* b3  = (const float*)d_in[8];   // [16]
  float* out = (float*)d_out;                 // [100000,16]

  // Workspace layout (102.4 MB total):
  float* h1  = (float*)d_ws;                          // 100000*64 f32
  float* h2  = h1 + (size_t)GIN_N_NODES * 64;         // 100000*64 f32
  float* agg = h2 + (size_t)GIN_N_NODES * 64;         // 100000*128 f32 (reused)

  const int M = GIN_N_NODES;
  const int mGroups = cdiv(M / 16, 8);                // 782 blocks of 8 M-tiles

  // ---- Layer 1: dim 128 -> 64 ----
  run_zero(agg, (long long)M * 128, stream);
  gin_scatter_kernel<5><<<cdiv((long long)GIN_N_EDGES * 32, 256), 256, 0, stream>>>(
      x, src, dst, agg, GIN_N_EDGES);
  gin_gemm_kernel<128><<<dim3(mGroups, 64 / 16), 256, 0, stream>>>(
      x, agg, w1, b1, h1, M, 64);

  // ---- Layer 2: dim 64 -> 64 ----
  run_zero(agg, (long long)M * 64, stream);
  gin_scatter_kernel<4><<<cdiv((long long)GIN_N_EDGES * 16, 256), 256, 0, stream>>>(
      h1, src, dst, agg, GIN_N_EDGES);
  gin_gemm_kernel<64><<<dim3(mGroups, 64 / 16), 256, 0, stream>>>(
      h1, agg, w2, b2, h2, M, 64);

  // ---- Layer 3: dim 64 -> 16 ----
  run_zero(agg, (long long)M * 64, stream);
  gin_scatter_kernel<4><<<cdiv((long long)GIN_N_EDGES * 16, 256), 256, 0, stream>>>(
      h2, src, dst, agg, GIN_N_EDGES);
  gin_gemm_kernel<64><<<dim3(mGroups, 16 / 16), 256, 0, stream>>>(
      h2, agg, w3, b3, out, M, 16);
}